// ImprovedGNNEncoder_v2_19550691131404
// MI455X (gfx1250) — compile-verified
//
#include <hip/hip_runtime.h>
#include <hip/hip_bf16.h>
#include <math.h>

// ImprovedGNNEncoder for MI455X (gfx1250).
// GEMMs use V_WMMA_F32_16X16X4_F32 (exact fp32 matrix path, wave32).
// Graph aggregation / edge-softmax use atomics; the 192MB L2 holds the
// full feature matrices (<=102MB) so scatter/gather hits L2, not HBM.

typedef float v2f __attribute__((ext_vector_type(2)));
typedef float v8f __attribute__((ext_vector_type(8)));

#define HIDC 64
#define NHEAD 4
#define NEG_SLOPE 0.2f
#define BN_EPS 1e-5f

// ---------------------------------------------------------------- utilities
__global__ void fill_f32(float* p, float v, long long n) {
  long long i = (long long)blockIdx.x * blockDim.x + threadIdx.x;
  long long stride = (long long)gridDim.x * blockDim.x;
  for (; i < n; i += stride) p[i] = v;
}

__device__ __forceinline__ void atomicMaxF(float* addr, float v) {
  if (v >= 0.0f) atomicMax((int*)addr, __float_as_int(v));
  else           atomicMin((unsigned int*)addr, __float_as_uint(v));
}

__device__ __forceinline__ void edge_nodes(const int* ei, long long E, long long e,
                                           int& s, int& d) {
  if (e < E) { s = ei[e]; d = ei[E + e]; }
  else       { s = (int)(e - E); d = s; }   // self loop
}

// ---------------------------------------------------------------- degree
__global__ void deg_accum(const int* __restrict__ ei, long long E, float* deg) {
  long long i = (long long)blockIdx.x * blockDim.x + threadIdx.x;
  long long stride = (long long)gridDim.x * blockDim.x;
  for (; i < E; i += stride) atomicAdd(&deg[ei[E + i]], 1.0f);
}

__global__ void dinv_k(float* deg, int n) {
  int i = blockIdx.x * blockDim.x + threadIdx.x;
  if (i < n) deg[i] = rsqrtf(fmaxf(deg[i], 1.0f));
}

// ---------------------------------------------------------------- WMMA GEMM
// C[nRowTiles*16 x Kout] = A[... x Kin] * W[Kin x Kout]
// One wave -> 16 rows x 64 cols.  fp32 WMMA 16x16x4.
// A layout : lane l: m=l&15, comp c + 2*(l>>4) selects K within the 4-slab.
// B layout : lane l: n=l&15, same K mapping.
// C layout : comp v, half h: M = v + 8h, N = l&15.
__global__ __launch_bounds__(256)
void gemm_wmma_f32(const float* __restrict__ A, const float* __restrict__ W,
                   float* __restrict__ C, int nRowTiles, int Kin, int Kout) {
  const int wave = threadIdx.x >> 5;
  const int lane = threadIdx.x & 31;
  const int rowTile = blockIdx.x * 8 + wave;
  if (rowTile >= nRowTiles) return;       // wave-uniform exit (EXEC stays full)
  const int half = lane >> 4;
  const int l16  = lane & 15;
  const int colBase = blockIdx.y * 64;
  const float* arow = A + (size_t)(rowTile * 16 + l16) * Kin;
  v8f acc0 = {}, acc1 = {}, acc2 = {}, acc3 = {};
  for (int k = 0; k < Kin; k += 4) {
    v2f a;
    a.x = arow[k + 2 * half];
    a.y = arow[k + 2 * half + 1];
    const float* w0 = W + (size_t)(k + 2 * half) * Kout + colBase + l16;
    v2f b0, b1, b2, b3;
    b0.x = w0[0];  b0.y = w0[Kout + 0];
    b1.x = w0[16]; b1.y = w0[Kout + 16];
    b2.x = w0[32]; b2.y = w0[Kout + 32];
    b3.x = w0[48]; b3.y = w0[Kout + 48];
    acc0 = __builtin_amdgcn_wmma_f32_16x16x4_f32(false, a, false, b0, (short)0, acc0, false, false);
    acc1 = __builtin_amdgcn_wmma_f32_16x16x4_f32(false, a, false, b1, (short)0, acc1, false, false);
    acc2 = __builtin_amdgcn_wmma_f32_16x16x4_f32(false, a, false, b2, (short)0, acc2, false, false);
    acc3 = __builtin_amdgcn_wmma_f32_16x16x4_f32(false, a, false, b3, (short)0, acc3, false, false);
  }
  for (int v = 0; v < 8; ++v) {
    int row = rowTile * 16 + v + 8 * half;
    float* crow = C + (size_t)row * Kout + colBase + l16;
    crow[0]  = acc0[v];
    crow[16] = acc1[v];
    crow[32] = acc2[v];
    crow[48] = acc3[v];
  }
}

// ---------------------------------------------------------------- GCN aggregation
__global__ __launch_bounds__(256)
void gcn_agg(const float* __restrict__ h, const int* __restrict__ ei,
             const float* __restrict__ dinv, float* __restrict__ out,
             long long E, int n) {
  long long Etot = E + n;
  long long e = (long long)blockIdx.x * 4 + (threadIdx.x >> 6);
  int c = threadIdx.x & 63;
  if (e >= Etot) return;
  int s, d; edge_nodes(ei, E, e, s, d);
  float norm = dinv[s] * dinv[d];
  atomicAdd(&out[(size_t)d * HIDC + c], h[(size_t)s * HIDC + c] * norm);
}

// ---------------------------------------------------------------- BatchNorm
// mode 0: stats over (x + b[c]);  mode 1: write relu(x+b) back, stats over it
__global__ __launch_bounds__(256)
void bn_stats(float* __restrict__ X, const float* __restrict__ bias,
              float* __restrict__ stats, int n, int C, int mode) {
  __shared__ float sm[512];
  int tid = threadIdx.x;
  int rowsPer = 256 / C;
  int c = tid % C;
  int r0 = tid / C;
  float b = bias ? bias[c] : 0.0f;
  float s = 0.f, q = 0.f;
  for (long long row = (long long)blockIdx.x * rowsPer + r0; row < n;
       row += (long long)gridDim.x * rowsPer) {
    float v = X[row * C + c] + b;
    if (mode == 1) { v = fmaxf(v, 0.f); X[row * C + c] = v; }
    s += v; q += v * v;
  }
  sm[tid] = s; sm[256 + tid] = q;
  __syncthreads();
  if (tid < C) {
    float ts = 0.f, tq = 0.f;
    for (int r = 0; r < rowsPer; ++r) { ts += sm[tid + r * C]; tq += sm[256 + tid + r * C]; }
    atomicAdd(&stats[tid], ts);
    atomicAdd(&stats[C + tid], tq);
  }
}

__global__ void bn_finalize(const float* __restrict__ stats, const float* __restrict__ g,
                            const float* __restrict__ beta, float* __restrict__ scsh,
                            int n, int C) {
  int c = blockIdx.x * blockDim.x + threadIdx.x;
  if (c >= C) return;
  float inv = 1.0f / (float)n;
  float mean = stats[c] * inv;
  float var  = stats[C + c] * inv - mean * mean;   // biased, matches jnp.var
  float sc = g[c] * rsqrtf(var + BN_EPS);
  scsh[c] = sc;
  scsh[C + c] = beta[c] - mean * sc;
}

// act 0: none; act 1: ELU(alpha=1). bias added pre-scale (null -> 0).
__global__ __launch_bounds__(256)
void bn_apply(float* __restrict__ X, const float* __restrict__ bias,
              const float* __restrict__ scsh, long long total, int C, int act) {
  long long i = (long long)blockIdx.x * blockDim.x + threadIdx.x;
  long long stride = (long long)gridDim.x * blockDim.x;
  for (; i < total; i += stride) {
    int c = (int)(i % C);
    float v = X[i] + (bias ? bias[c] : 0.f);
    v = v * scsh[c] + scsh[C + c];
    if (act == 1) v = v > 0.f ? v : expm1f(v);
    X[i] = v;
  }
}

// ---------------------------------------------------------------- GAT attention
// one wave per (node, head): two 64-long dot products
__global__ __launch_bounds__(256)
void gat_alpha(const float* __restrict__ h, const float* __restrict__ a_src,
               const float* __restrict__ a_dst, float* __restrict__ as_,
               float* __restrict__ ad_, int n) {
  int lane = threadIdx.x & 31;
  long long w = (long long)blockIdx.x * 8 + (threadIdx.x >> 5);
  if (w >= (long long)n * NHEAD) return;
  int node = (int)(w >> 2);
  int head = (int)(w & 3);
  const float* hp  = h + (size_t)node * (NHEAD * HIDC) + head * HIDC;
  const float* asp = a_src + head * HIDC;
  const float* adp = a_dst + head * HIDC;
  float ss = hp[lane] * asp[lane] + hp[lane + 32] * asp[lane + 32];
  float sd = hp[lane] * adp[lane] + hp[lane + 32] * adp[lane + 32];
  for (int o = 16; o > 0; o >>= 1) {
    ss += __shfl_xor(ss, o, 32);
    sd += __shfl_xor(sd, o, 32);
  }
  if (lane == 0) { as_[w] = ss; ad_[w] = sd; }
}

__global__ __launch_bounds__(256)
void att_pass1(const int* __restrict__ ei, const float* __restrict__ as_,
               const float* __restrict__ ad_, float* __restrict__ wbuf,
               float* __restrict__ emax, long long E, int n) {
  long long tot = (E + n) * NHEAD;
  long long i = (long long)blockIdx.x * blockDim.x + threadIdx.x;
  long long stride = (long long)gridDim.x * blockDim.x;
  for (; i < tot; i += stride) {
    long long e = i >> 2; int hh = (int)(i & 3);
    int s, d; edge_nodes(ei, E, e, s, d);
    float v = as_[(size_t)s * NHEAD + hh] + ad_[(size_t)d * NHEAD + hh];
    v = v > 0.f ? v : NEG_SLOPE * v;          // LeakyReLU
    wbuf[i] = v;
    atomicMaxF(&emax[(size_t)d * NHEAD + hh], v);
  }
}

__global__ __launch_bounds__(256)
void att_pass2(const int* __restrict__ ei, float* __restrict__ wbuf,
               const float* __restrict__ emax, float* __restrict__ wsum,
               long long E, int n) {
  long long tot = (E + n) * NHEAD;
  long long i = (long long)blockIdx.x * blockDim.x + threadIdx.x;
  long long stride = (long long)gridDim.x * blockDim.x;
  for (; i < tot; i += stride) {
    long long e = i >> 2; int hh = (int)(i & 3);
    int d = (e < E) ? ei[E + e] : (int)(e - E);
    float w = __expf(wbuf[i] - emax[(size_t)d * NHEAD + hh]);
    wbuf[i] = w;
    atomicAdd(&wsum[(size_t)d * NHEAD + hh], w);
  }
}

__global__ __launch_bounds__(256)
void att_scatter(const float* __restrict__ h, const int* __restrict__ ei,
                 const float* __restrict__ wbuf, const float* __restrict__ wsum,
                 float* __restrict__ out, long long E, int n) {
  long long e = blockIdx.x;
  if (e >= E + n) return;
  int head = threadIdx.x >> 6;
  int c    = threadIdx.x & 63;
  int s, d; edge_nodes(ei, E, e, s, d);
  float att = wbuf[e * NHEAD + head] / wsum[(size_t)d * NHEAD + head];
  atomicAdd(&out[(size_t)d * (NHEAD * HIDC) + head * HIDC + c],
            att * h[(size_t)s * (NHEAD * HIDC) + head * HIDC + c]);
}

// ---------------------------------------------------------------- launch
extern "C" void kernel_launch(void* const* d_in, const int* in_sizes, int n_in,
                              void* d_out, int out_size, void* d_ws, size_t ws_size,
                              hipStream_t stream) {
  const float* x      = (const float*)d_in[0];
  const int*   ei     = (const int*)d_in[1];
  const float* W_gcn  = (const float*)d_in[2];
  const float* b_gcn  = (const float*)d_in[3];
  const float* g0     = (const float*)d_in[4];
  const float* beta0  = (const float*)d_in[5];
  const float* W1     = (const float*)d_in[6];
  const float* a_src1 = (const float*)d_in[7];
  const float* a_dst1 = (const float*)d_in[8];
  const float* b1     = (const float*)d_in[9];
  const float* g1     = (const float*)d_in[10];
  const float* beta1  = (const float*)d_in[11];
  const float* W2     = (const float*)d_in[12];
  const float* a_src2 = (const float*)d_in[13];
  const float* a_dst2 = (const float*)d_in[14];
  const float* b2     = (const float*)d_in[15];
  const float* g2     = (const float*)d_in[16];
  const float* beta2  = (const float*)d_in[17];
  float* out = (float*)d_out;

  const int       Nn   = in_sizes[0] / 256;          // 100000
  const long long E    = in_sizes[1] / 2;            // 800000
  const long long Etot = E + Nn;
  const int nRowTiles  = Nn / 16;                    // N is a multiple of 16

  // ---- workspace layout (floats) -----------------------------------------
  float* f = (float*)d_ws;
  size_t off = 0;
  float* dinv = f + off; off += (size_t)Nn;
  float* B0   = f + off; off += (size_t)Nn * 64;     // h = x @ W_gcn
  float* Bagg = f + off; off += (size_t)Nn * 64;     // gcn agg -> bn0 out
  float* B2   = f + off; off += (size_t)Nn * 256;    // h1 (later reused as h2)
  float* B3   = f + off; off += (size_t)Nn * 256;    // gat1 out -> bn1/elu
  float* as_  = f + off; off += (size_t)Nn * NHEAD;
  float* ad_  = f + off; off += (size_t)Nn * NHEAD;
  float* emax = f + off; off += (size_t)Nn * NHEAD;
  float* wsum = f + off; off += (size_t)Nn * NHEAD;
  float* wbuf = f + off; off += (size_t)Etot * NHEAD;
  float* stats = f + off; off += 512;
  float* scsh  = f + off; off += 512;
  (void)ws_size; (void)n_in; (void)out_size;

  const int TB = 256;
  // ---- degree / norm -----------------------------------------------------
  fill_f32<<<1024, TB, 0, stream>>>(dinv, 1.0f, Nn);           // self-loop deg
  deg_accum<<<1024, TB, 0, stream>>>(ei, E, dinv);
  dinv_k<<<(Nn + TB - 1) / TB, TB, 0, stream>>>(dinv, Nn);

  // ---- layer 0: GCN ------------------------------------------------------
  gemm_wmma_f32<<<dim3((nRowTiles + 7) / 8, 1), TB, 0, stream>>>(x, W_gcn, B0,
                                                                 nRowTiles, 256, 64);
  fill_f32<<<2048, TB, 0, stream>>>(Bagg, 0.f, (long long)Nn * 64);
  gcn_agg<<<(unsigned)((Etot + 3) / 4), TB, 0, stream>>>(B0, ei, dinv, Bagg, E, Nn);
  fill_f32<<<1, TB, 0, stream>>>(stats, 0.f, 512);
  bn_stats<<<1024, TB, 0, stream>>>(Bagg, b_gcn, stats, Nn, 64, 1);   // relu+stats
  bn_finalize<<<1, TB, 0, stream>>>(stats, g0, beta0, scsh, Nn, 64);
  bn_apply<<<2048, TB, 0, stream>>>(Bagg, nullptr, scsh, (long long)Nn * 64, 64, 0);

  // ---- layer 1: GAT ------------------------------------------------------
  gemm_wmma_f32<<<dim3((nRowTiles + 7) / 8, 4), TB, 0, stream>>>(Bagg, W1, B2,
                                                                 nRowTiles, 64, 256);
  gat_alpha<<<(unsigned)(((long long)Nn * NHEAD + 7) / 8), TB, 0, stream>>>(
      B2, a_src1, a_dst1, as_, ad_, Nn);
  fill_f32<<<1024, TB, 0, stream>>>(emax, -3.0e38f, (long long)Nn * NHEAD);
  fill_f32<<<1024, TB, 0, stream>>>(wsum, 0.f, (long long)Nn * NHEAD);
  att_pass1<<<2048, TB, 0, stream>>>(ei, as_, ad_, wbuf, emax, E, Nn);
  att_pass2<<<2048, TB, 0, stream>>>(ei, wbuf, emax, wsum, E, Nn);
  fill_f32<<<4096, TB, 0, stream>>>(B3, 0.f, (long long)Nn * 256);
  att_scatter<<<(unsigned)Etot, TB, 0, stream>>>(B2, ei, wbuf, wsum, B3, E, Nn);
  fill_f32<<<1, TB, 0, stream>>>(stats, 0.f, 512);
  bn_stats<<<1024, TB, 0, stream>>>(B3, b1, stats, Nn, 256, 0);
  bn_finalize<<<1, TB, 0, stream>>>(stats, g1, beta1, scsh, Nn, 256);
  bn_apply<<<4096, TB, 0, stream>>>(B3, b1, scsh, (long long)Nn * 256, 256, 1); // elu

  // ---- layer 2: GAT ------------------------------------------------------
  gemm_wmma_f32<<<dim3((nRowTiles + 7) / 8, 4), TB, 0, stream>>>(B3, W2, B2,
                                                                 nRowTiles, 256, 256);
  gat_alpha<<<(unsigned)(((long long)Nn * NHEAD + 7) / 8), TB, 0, stream>>>(
      B2, a_src2, a_dst2, as_, ad_, Nn);
  fill_f32<<<1024, TB, 0, stream>>>(emax, -3.0e38f, (long long)Nn * NHEAD);
  fill_f32<<<1024, TB, 0, stream>>>(wsum, 0.f, (long long)Nn * NHEAD);
  att_pass1<<<2048, TB, 0, stream>>>(ei, as_, ad_, wbuf, emax, E, Nn);
  att_pass2<<<2048, TB, 0, stream>>>(ei, wbuf, emax, wsum, E, Nn);
  fill_f32<<<4096, TB, 0, stream>>>(out, 0.f, (long long)Nn * 256);
  att_scatter<<<(unsigned)Etot, TB, 0, stream>>>(B2, ei, wbuf, wsum, out, E, Nn);
  fill_f32<<<1, TB, 0, stream>>>(stats, 0.f, 512);
  bn_stats<<<1024, TB, 0, stream>>>(out, b2, stats, Nn, 256, 0);
  bn_finalize<<<1, TB, 0, stream>>>(stats, g2, beta2, scsh, Nn, 256);
  bn_apply<<<4096, TB, 0, stream>>>(out, b2, scsh, (long long)Nn * 256, 256, 1); // elu
}